// GaussianRenderer_11218454577223
// MI455X (gfx1250) — compile-verified
//
#include <hip/hip_runtime.h>
#include <stdint.h>

#define TPB 256
#define SCAN_TILE 2048   // 8 waves * 256 elems (one 16x16 f32 tile per wave)

typedef float v2f __attribute__((ext_vector_type(2)));
typedef float v8f __attribute__((ext_vector_type(8)));

// ---- order-preserving float <-> uint key (for atomic min/max on floats) ----
__device__ __forceinline__ unsigned fkey(float f){
  unsigned u = __float_as_uint(f);
  return (u & 0x80000000u) ? ~u : (u | 0x80000000u);
}
__device__ __forceinline__ float kfloat(unsigned k){
  unsigned u = (k & 0x80000000u) ? (k ^ 0x80000000u) : ~k;
  return __uint_as_float(u);
}

struct Rec { unsigned idx; float a, r, g, b, z; };  // 24 B

// ---------------- init: minmax keys + zero histogram ----------------
__global__ void k_init(unsigned* __restrict__ mm, unsigned* __restrict__ cnt, int n){
  int i = blockIdx.x * blockDim.x + threadIdx.x;
  if (i < n) cnt[i] = 0u;
  if (i == 0){ mm[0] = 0xFFFFFFFFu; mm[1] = 0u; mm[2] = 0xFFFFFFFFu; mm[3] = 0u; }
}

// ---------------- min/max of x,y; streams xyz via async global->LDS ----------------
__global__ void k_minmax(const float* __restrict__ xyz, unsigned* __restrict__ mm, int nf){
  // per-wave double-buffered staging: 2 bufs * 8 waves * 32 lanes * 16B = 8 KB
  volatile __shared__ float sh[2][8][128];
  if (nf <= 0) return;
  const int  lane = threadIdx.x & 31;
  const int  wIn  = threadIdx.x >> 5;
  const long gw   = (long)blockIdx.x * (blockDim.x >> 5) + wIn;
  const long NW   = (long)gridDim.x * (blockDim.x >> 5);
  const long nf4  = ((long)nf + 3) >> 2;
  const long iters = (nf4 + NW * 32 - 1) / (NW * 32);

  float mnx =  INFINITY, mxx = -INFINITY, mny =  INFINITY, mxy = -INFINITY;

  auto issue = [&](long itv, int buf){
    long f4 = (itv * NW + gw) * 32 + lane;
    if (f4 >= nf4) f4 = nf4 - 1;                  // clamp: keep all lanes active & uniform
    unsigned long long ga = (unsigned long long)(uintptr_t)((const char*)xyz + f4 * 16);
    unsigned lo = (unsigned)(uintptr_t)(&sh[buf][wIn][lane * 4]);
    asm volatile("global_load_async_to_lds_b128 %0, %1, off" :: "v"(lo), "v"(ga) : "memory");
  };

  issue(0, 0);
  for (long it = 0; it < iters; ++it){
    int buf = (int)(it & 1);
    if (it + 1 < iters){
      issue(it + 1, (int)((it + 1) & 1));
      asm volatile("s_wait_asynccnt 0x1" ::: "memory");   // oldest load done (in-order)
    } else {
      asm volatile("s_wait_asynccnt 0x0" ::: "memory");
    }
    long f4 = (it * NW + gw) * 32 + lane;
    if (f4 >= nf4) f4 = nf4 - 1;
    #pragma unroll
    for (int j = 0; j < 4; ++j){
      long f = f4 * 4 + j;                         // flat float index into (N,3)
      if (f < (long)nf){
        float v = sh[buf][wIn][lane * 4 + j];
        int r = (int)(f % 3);                      // 0=x, 1=y, 2=z(skip)
        if (r == 0){ mnx = fminf(mnx, v); mxx = fmaxf(mxx, v); }
        else if (r == 1){ mny = fminf(mny, v); mxy = fmaxf(mxy, v); }
      }
    }
  }
  // wave32 butterfly reduction, then one atomic per value per wave
  #pragma unroll
  for (int m = 16; m > 0; m >>= 1){
    mnx = fminf(mnx, __shfl_xor(mnx, m, 32));
    mxx = fmaxf(mxx, __shfl_xor(mxx, m, 32));
    mny = fminf(mny, __shfl_xor(mny, m, 32));
    mxy = fmaxf(mxy, __shfl_xor(mxy, m, 32));
  }
  if (lane == 0){
    atomicMin(&mm[0], fkey(mnx));
    atomicMax(&mm[1], fkey(mxx));
    atomicMin(&mm[2], fkey(mny));
    atomicMax(&mm[3], fkey(mxy));
  }
}

// ---------------- bin points to pixels, build per-pixel histogram ----------------
__global__ void k_bin(const float* __restrict__ xyz, const unsigned* __restrict__ mm,
                      unsigned* __restrict__ cnt, unsigned* __restrict__ pid,
                      int N, int W, int H){
  int i = blockIdx.x * blockDim.x + threadIdx.x;
  if (i >= N) return;
  float mnx = kfloat(mm[0]), mxx = kfloat(mm[1]);
  float mny = kfloat(mm[2]), mxy = kfloat(mm[3]);
  float x = xyz[3 * i], y = xyz[3 * i + 1];
  float nx = (x - mnx) / ((mxx - mnx) + 1e-8f);    // identical fp32 math to reference
  float ny = (y - mny) / ((mxy - mny) + 1e-8f);
  int xi = (int)(nx * (float)W);                    // trunc == astype(int32)
  int yi = (int)(ny * (float)H);
  bool valid = (xi >= 0) && (xi < W) && (yi >= 0) && (yi < H);
  unsigned p = valid ? (unsigned)(yi * W + xi) : 0xFFFFFFFFu;
  pid[i] = p;
  if (valid) atomicAdd(&cnt[p], 1u);
}

// ---------------- level-1 scan: WMMA prefix sums ----------------
// Each wave scans a 256-elem tile viewed as 16x16 row-major M.
// Exclusive within-row prefix = M x U'  (U'[i][j] = 1 iff i<j)
// Row sums (replicated)       = M x 1
// Both via 4 chained V_WMMA_F32_16X16X4_F32 each (K=16 split into 4xK=4).
// Exact: all partials are integers < 2^24, f32 lossless.
__global__ void k_scan1(unsigned* __restrict__ cnt, unsigned* __restrict__ offs,
                        unsigned* __restrict__ bsum, int n){
  __shared__ float wsum[TPB / 32];
  const int lane = threadIdx.x & 31;
  const int w    = threadIdx.x >> 5;
  const int hf   = lane >> 4;            // lane half: selects K-pair (A) / K-rows (B)
  const int r    = lane & 15;            // A-operand M-row; also D-operand column N
  const long base = (long)blockIdx.x * SCAN_TILE + (long)w * 256;

  // Load A_k operands straight from global in A layout:
  //   lanes 0-15 : M=r, v0=K(4k+0), v1=K(4k+1)
  //   lanes 16-31: M=r, v0=K(4k+2), v1=K(4k+3)
  v2f A[4];
  #pragma unroll
  for (int k = 0; k < 4; ++k){
    long i0 = base + (long)r * 16 + 4 * k + 2 * hf;
    A[k].x = (i0     < n) ? (float)cnt[i0]     : 0.f;
    A[k].y = (i0 + 1 < n) ? (float)cnt[i0 + 1] : 0.f;
  }

  v8f P = {};   // exclusive row prefix (D layout: row j+8*hf, col r)
  v8f S = {};   // row sums replicated across columns
  #pragma unroll
  for (int k = 0; k < 4; ++k){
    // B layout (4x16 f32): lanes 0-15 hold K rows 4k+0/4k+1, lanes 16-31 rows 4k+2/4k+3; N=lane&15
    int kk = 4 * k + 2 * hf;
    v2f Bu, Bo;
    Bu.x = (kk + 0 < r) ? 1.f : 0.f;     // strict upper-tri ones -> exclusive prefix
    Bu.y = (kk + 1 < r) ? 1.f : 0.f;
    Bo.x = 1.f; Bo.y = 1.f;              // all-ones -> row sums
    P = __builtin_amdgcn_wmma_f32_16x16x4_f32(false, A[k], false, Bu, (short)0, P, false, false);
    S = __builtin_amdgcn_wmma_f32_16x16x4_f32(false, A[k], false, Bo, (short)0, S, false, false);
  }

  // Gather all 16 row sums into every lane (other half via xor-16 shuffle)
  float s[16];
  #pragma unroll
  for (int j = 0; j < 8; ++j){
    float mine  = S[j];                          // row j + 8*hf (any column)
    float other = __shfl_xor(mine, 16, 32);      // row j + 8*(1-hf)
    s[j]     = (hf == 0) ? mine  : other;
    s[j + 8] = (hf == 0) ? other : mine;
  }
  // Serial exclusive scan of 16 row sums (every lane; 15 adds)
  float cum[16];
  cum[0] = 0.f;
  #pragma unroll
  for (int j = 1; j < 16; ++j) cum[j] = cum[j - 1] + s[j - 1];
  float waveTotal = cum[15] + s[15];

  // Cross-wave exclusive offsets via LDS
  if (lane == 0) wsum[w] = waveTotal;
  __syncthreads();
  float waveOff = 0.f;
  #pragma unroll
  for (int j = 0; j < TPB / 32; ++j) waveOff += (j < w) ? wsum[j] : 0.f;
  if (threadIdx.x == 0){
    float tot = 0.f;
    #pragma unroll
    for (int j = 0; j < TPB / 32; ++j) tot += wsum[j];
    bsum[blockIdx.x] = (unsigned)(tot + 0.5f);
  }

  // Write exclusive prefix (D layout) and re-zero cnt (reused as scatter cursor)
  #pragma unroll
  for (int j = 0; j < 8; ++j){
    int row = j + 8 * hf;
    float cj = (hf == 0) ? cum[j] : cum[j + 8];
    long idx = base + (long)row * 16 + r;
    if (idx < n){
      offs[idx] = (unsigned)(P[j] + cj + waveOff + 0.5f);
      cnt[idx]  = 0u;
    }
  }
}

__global__ void k_scan2(unsigned* __restrict__ b, int n){  // single block, n <= 2048
  __shared__ unsigned s[TPB];
  int t = threadIdx.x;
  int base = t * 8;
  unsigned v[8]; unsigned tsum = 0;
  #pragma unroll
  for (int j = 0; j < 8; ++j){ int idx = base + j; unsigned x = (idx < n) ? b[idx] : 0u; v[j] = x; tsum += x; }
  s[t] = tsum; __syncthreads();
  for (int o = 1; o < TPB; o <<= 1){
    unsigned x = (t >= o) ? s[t - o] : 0u;
    __syncthreads();
    s[t] += x;
    __syncthreads();
  }
  unsigned run = (t > 0) ? s[t - 1] : 0u;
  #pragma unroll
  for (int j = 0; j < 8; ++j){ int idx = base + j; if (idx < n) b[idx] = run; run += v[j]; }
}

__global__ void k_scan3(unsigned* __restrict__ offs, const unsigned* __restrict__ bsum, int n){
  int i = blockIdx.x * blockDim.x + threadIdx.x;
  if (i < n) offs[i] += bsum[i >> 11];   // SCAN_TILE == 2048
}

// ---------------- scatter per-point records into per-pixel segments ----------------
__global__ void k_scatter(const float* __restrict__ xyz, const float* __restrict__ feat,
                          const float* __restrict__ opac, const unsigned* __restrict__ pid,
                          const unsigned* __restrict__ offs, unsigned* __restrict__ cur,
                          Rec* __restrict__ recs, int N, int featStride){
  int i = blockIdx.x * blockDim.x + threadIdx.x;
  if (i >= N) return;
  unsigned p = pid[i];
  if (p == 0xFFFFFFFFu) return;
  unsigned slot = offs[p] + atomicAdd(&cur[p], 1u);
  long fb = (long)i * featStride;                   // features[:,0,:] = first 3 of K*3
  Rec rc;
  rc.idx = (unsigned)i;
  rc.a   = opac[i];
  rc.r   = fminf(fmaxf(feat[fb + 0] + 0.5f, 0.f), 1.f);
  rc.g   = fminf(fmaxf(feat[fb + 1] + 0.5f, 0.f), 1.f);
  rc.b   = fminf(fmaxf(feat[fb + 2] + 0.5f, 0.f), 1.f);
  rc.z   = xyz[3 * i + 2];
  recs[slot] = rc;
}

// ---------------- per-pixel exact composite in ascending index order ----------------
__global__ void k_composite(const Rec* __restrict__ recs, const unsigned* __restrict__ offs,
                            const float* __restrict__ bg, float* __restrict__ out, int HW){
  int p = blockIdx.x * blockDim.x + threadIdx.x;
  if (p >= HW) return;
  unsigned s0 = offs[p], s1 = offs[p + 1];
  float T = 1.f, Cr = 0.f, Cg = 0.f, Cb = 0.f, depth = 0.f;
  long lastIdx = -1;
  for (unsigned k = s0; k < s1; ++k){               // selection-walk: min idx > lastIdx
    unsigned best = 0xFFFFFFFFu, bs = s0;
    for (unsigned s = s0; s < s1; ++s){
      unsigned id = recs[s].idx;
      if ((long)id > lastIdx && id < best){ best = id; bs = s; }
    }
    Rec rc = recs[bs];
    float om = 1.f - rc.a;
    Cr = Cr * om + rc.a * rc.r;
    Cg = Cg * om + rc.a * rc.g;
    Cb = Cb * om + rc.a * rc.b;
    T *= om;
    depth = rc.z;                                    // last applied = max index -> ref depth
    lastIdx = (long)best;
  }
  float b0 = bg[0], b1 = bg[1], b2 = bg[2];
  // write-once 41.5 MB output: non-temporal stores, keep L2 for scratch
  __builtin_nontemporal_store(b0 * T + Cr, out + 0L * HW + p);
  __builtin_nontemporal_store(b1 * T + Cg, out + 1L * HW + p);
  __builtin_nontemporal_store(b2 * T + Cb, out + 2L * HW + p);
  __builtin_nontemporal_store(depth,       out + 3L * HW + p);
  __builtin_nontemporal_store(1.f - T,     out + 4L * HW + p);
}

extern "C" void kernel_launch(void* const* d_in, const int* in_sizes, int n_in,
                              void* d_out, int out_size, void* d_ws, size_t ws_size,
                              hipStream_t stream){
  const float* xyz  = (const float*)d_in[0];
  const float* feat = (const float*)d_in[1];
  const float* opac = (const float*)d_in[2];
  const float* bg   = (const float*)d_in[3];
  const int N          = in_sizes[0] / 3;
  const int featStride = (N > 0) ? in_sizes[1] / N : 48;  // K*3
  const int HW         = out_size / 5;                    // 3 color + depth + alpha planes
  const int W          = 1920;                            // reference module constants
  const int H          = (W > 0) ? HW / W : 0;
  const int nScan      = HW + 1;
  const int NB1        = (nScan + SCAN_TILE - 1) / SCAN_TILE;

  char* ws = (char*)d_ws;
  size_t off = 0;
  auto carve = [&](size_t bytes){ size_t o = off; off = (off + bytes + 255) & ~(size_t)255; return o; };
  unsigned* mm   = (unsigned*)(ws + carve(4 * sizeof(unsigned)));
  unsigned* cnt  = (unsigned*)(ws + carve((size_t)nScan * 4));
  unsigned* offs = (unsigned*)(ws + carve((size_t)nScan * 4));
  unsigned* bsum = (unsigned*)(ws + carve((size_t)NB1 * 4));
  unsigned* pid  = (unsigned*)(ws + carve((size_t)N * 4));
  Rec*      recs = (Rec*)     (ws + carve((size_t)N * sizeof(Rec)));
  (void)ws_size; (void)n_in;

  k_init     <<<(nScan + TPB - 1) / TPB, TPB, 0, stream>>>(mm, cnt, nScan);
  k_minmax   <<<256, TPB, 0, stream>>>(xyz, mm, 3 * N);
  k_bin      <<<(N + TPB - 1) / TPB, TPB, 0, stream>>>(xyz, mm, cnt, pid, N, W, H);
  k_scan1    <<<NB1, TPB, 0, stream>>>(cnt, offs, bsum, nScan);
  k_scan2    <<<1, TPB, 0, stream>>>(bsum, NB1);
  k_scan3    <<<(nScan + TPB - 1) / TPB, TPB, 0, stream>>>(offs, bsum, nScan);
  k_scatter  <<<(N + TPB - 1) / TPB, TPB, 0, stream>>>(xyz, feat, opac, pid, offs, cnt, recs, N, featStride);
  k_composite<<<(HW + TPB - 1) / TPB, TPB, 0, stream>>>(recs, offs, bg, (float*)d_out, HW);
}